// RNNT_68229850464312
// MI455X (gfx1250) — compile-verified
//
#include <hip/hip_runtime.h>

typedef float v2f __attribute__((ext_vector_type(2)));
typedef float v8f __attribute__((ext_vector_type(8)));

// D = A(16x4 f32) * B(4x16 f32) + C(16x16 f32), full fp32 matrix op (CDNA5 WMMA)
#define WMMA_F32(a, b, acc) \
  __builtin_amdgcn_wmma_f32_16x16x4_f32(false, (a), false, (b), (short)0, (acc), false, false)

__device__ __forceinline__ float sigm(float x) { return 1.0f / (1.0f + __expf(-x)); }

// ---------------------------------------------------------------------------
// Cooperative gate computation for one recurrent step:
//   gS[4][2048] = (Gin rows, stride 2048) + bias + hS[4][512] @ Wh[2048][512]^T
//                 (+ X[4][Kx] @ Wx[2048][Kx]^T  if X != nullptr)
// Block = 1024 threads = 32 waves; each wave owns 4 of the 128 16-wide tiles.
// A-rows 4..15 are zero-filled via clamp+select (EXEC stays all-ones for WMMA).
// ---------------------------------------------------------------------------
__device__ __forceinline__ void gates_step(
    const float* __restrict__ Gin, const float* __restrict__ bias,
    const float* __restrict__ Wh, const float* __restrict__ hS,
    const float* __restrict__ X, const float* __restrict__ Wx, int Kx,
    float* __restrict__ gS, int tid)
{
  const int wave  = tid >> 5;
  const int lane  = tid & 31;
  const int m     = lane & 15;
  const int khalf = (lane >> 4) << 1;   // lanes 0-15 -> K {0,1}; lanes 16-31 -> K {2,3}
  const int hm    = (m < 4) ? m : 0;

  for (int i = 0; i < 4; ++i) {
    const int nt   = wave + (i << 5);       // tile 0..127
    const int ncol = (nt << 4) + m;         // weight row (gate column)
    const float* wrow = Wh + (size_t)ncol * 512;
    const float* hrow = hS + hm * 512;
    v8f acc = {};
    for (int k = 0; k < 512; k += 4) {
      v2f a = *(const v2f*)(hrow + k + khalf);
      if (m >= 4) { a.x = 0.f; a.y = 0.f; }
      v2f b = *(const v2f*)(wrow + k + khalf);
      acc = WMMA_F32(a, b, acc);
    }
    if (X) {
      const float* xrow  = X  + (size_t)hm * Kx;
      const float* wxrow = Wx + (size_t)ncol * Kx;
      for (int k = 0; k < Kx; k += 4) {
        v2f a = *(const v2f*)(xrow + k + khalf);
        if (m >= 4) { a.x = 0.f; a.y = 0.f; }
        v2f b = *(const v2f*)(wxrow + k + khalf);
        acc = WMMA_F32(a, b, acc);
      }
    }
    // C/D layout: VGPR r, lanes<16 -> M=r.  Only rows 0..3 (batch) are valid.
    if (lane < 16) {
      const int col = (nt << 4) + lane;
      float base = bias ? bias[col] : 0.f;
#pragma unroll
      for (int r = 0; r < 4; ++r) {
        float v = acc[r] + base;
        if (Gin) v += Gin[r * 2048 + col];
        gS[r * 2048 + col] = v;
      }
    }
  }
}

// LSTM cell elementwise update: 4 batch x 512 hidden over 1024 threads.
__device__ __forceinline__ void cell_step(
    const float* __restrict__ gS, float* __restrict__ hS, float* __restrict__ cS,
    float* __restrict__ out, int outStride, int colOff, int tid)
{
  for (int idx = tid; idx < 2048; idx += 1024) {
    int b = idx >> 9;
    int j = idx & 511;
    const float* gb = gS + (b << 11);
    float ig = sigm(gb[j]);
    float fg = sigm(gb[512 + j]);
    float gg = tanhf(gb[1024 + j]);
    float og = sigm(gb[1536 + j]);
    float c = fg * cS[idx] + ig * gg;
    float h = og * tanhf(c);
    cS[idx] = c;
    hS[idx] = h;
    out[(size_t)b * outStride + colOff + j] = h;
  }
}

// ---------------------------------------------------------------------------
// Encoder recurrence: one persistent block per direction, loops over T steps.
// G: (2, T, 4, 2048) precomputed x@Wi.T + b.  out: (T, 4, 1024), col off dir*512.
// ---------------------------------------------------------------------------
__global__ void __launch_bounds__(1024)
enc_rec_kernel(const float* __restrict__ G, const float* __restrict__ Wh,
               float* __restrict__ out, int T)
{
  __shared__ float hS[2048], cS[2048], gS[8192];   // 48 KB
  const int dir = blockIdx.x;
  const int tid = threadIdx.x;
  const float* Whd = Wh + (size_t)dir * 1048576;   // 2048*512
  for (int i = tid; i < 2048; i += 1024) { hS[i] = 0.f; cS[i] = 0.f; }
  __syncthreads();
  for (int s = 0; s < T; ++s) {
    int t = dir ? (T - 1 - s) : s;
    gates_step(G + ((size_t)dir * T + t) * 8192, nullptr, Whd, hS,
               nullptr, nullptr, 0, gS, tid);
    __syncthreads();
    cell_step(gS, hS, cS, out + (size_t)t * 4096, 1024, dir * 512, tid);
    __syncthreads();
  }
}

// ---------------------------------------------------------------------------
// Decoder: single persistent block fusing both bidir layers over 65 scan steps.
// Dynamic LDS (147456 B, gfx1250 allows 320KB/WGP):
//   st[16384]: (h,c) for l0f,l0b,l1f,l1b   gS[8192]: gates   y0[12288]: (3,4,1024)
// G0: (2, 784, 2048) precomputed emb@Wi0.T + b0.  y1: (784, 1024) global out.
// ---------------------------------------------------------------------------
__global__ void __launch_bounds__(1024)
dec_rec_kernel(const float* __restrict__ G0, const float* __restrict__ Wh0,
               const float* __restrict__ Wi1, const float* __restrict__ Wh1,
               const float* __restrict__ b1, float* __restrict__ y1)
{
  extern __shared__ float sm[];
  float* st = sm;             // 4 * (h 2048 + c 2048)
  float* gS = sm + 16384;
  float* y0 = sm + 24576;     // (3 positions, 4 batch, 1024)
  const int tid = threadIdx.x;
  for (int i = tid; i < 16384; i += 1024) st[i] = 0.f;
  __syncthreads();

  for (int s = 0; s < 65; ++s) {
    // layer 0: fwd scans p=0,1,2 ; bwd scans p=2,1,0 (states carried across s)
    for (int q = 0; q < 3; ++q)
      for (int d = 0; d < 2; ++d) {
        int p = d ? (2 - q) : q;
        float* h = st + d * 4096;
        float* c = h + 2048;
        gates_step(G0 + ((size_t)d * 784 + s * 12 + p * 4) * 2048, nullptr,
                   Wh0 + (size_t)d * 1048576, h, nullptr, nullptr, 0, gS, tid);
        __syncthreads();
        cell_step(gS, h, c, y0 + p * 4096, 1024, d * 512, tid);
        __syncthreads();
      }
    // layer 1: input projection y0@Wi1.T fused into the step (K=1024)
    for (int q = 0; q < 3; ++q)
      for (int d = 0; d < 2; ++d) {
        int p = d ? (2 - q) : q;
        float* h = st + 8192 + d * 4096;
        float* c = h + 2048;
        gates_step(nullptr, b1 + d * 2048, Wh1 + (size_t)d * 1048576, h,
                   y0 + p * 4096, Wi1 + (size_t)d * 2097152, 1024, gS, tid);
        __syncthreads();
        cell_step(gS, h, c, y1 + ((size_t)s * 12 + p * 4) * 1024, 1024, d * 512, tid);
        __syncthreads();
      }
  }
}

// ---------------------------------------------------------------------------
// Generic WMMA GEMM: Y[M][N] = A[M][K] @ W[N][K]^T (+ bias).  256 thr = 8 waves,
// one 16x16 tile per wave.  Clamped loads + guarded stores for ragged M/N.
// ---------------------------------------------------------------------------
__global__ void __launch_bounds__(256)
gemm_wmma_kernel(const float* __restrict__ A, const float* __restrict__ W,
                 const float* __restrict__ bias, float* __restrict__ Y,
                 int M, int N, int K, int ldY)
{
  const int wave = threadIdx.x >> 5;
  const int lane = threadIdx.x & 31;
  const int Mt = (M + 15) >> 4;
  const int Nt = (N + 15) >> 4;
  const int tile = blockIdx.x * 8 + wave;
  if (tile >= Mt * Nt) return;             // wave-uniform exit
  const int mt = tile / Nt, nt = tile % Nt;
  const int m = lane & 15;
  const int khalf = (lane >> 4) << 1;
  int arow = (mt << 4) + m; if (arow > M - 1) arow = M - 1;
  int wrow = (nt << 4) + m; if (wrow > N - 1) wrow = N - 1;
  const float* Ar = A + (size_t)arow * K;
  const float* Wr = W + (size_t)wrow * K;
  v8f acc = {};
  for (int k = 0; k < K; k += 4) {
    v2f a = *(const v2f*)(Ar + k + khalf);
    v2f b = *(const v2f*)(Wr + k + khalf);
    acc = WMMA_F32(a, b, acc);
  }
  const int col = (nt << 4) + m;
  const int rbase = (mt << 4) + ((lane >> 4) << 3);
  if (col < N) {
    float bv = bias ? bias[col] : 0.f;
#pragma unroll
    for (int r = 0; r < 8; ++r) {
      int row = rbase + r;
      if (row < M) Y[(size_t)row * ldY + col] = acc[r] + bv;
    }
  }
}

// signal (4,200,80) -> xs (800,80) rows = t*4+b
__global__ void transpose_signal_kernel(const float* __restrict__ sig,
                                        float* __restrict__ xs)
{
  int idx = blockIdx.x * 256 + threadIdx.x;
  if (idx >= 800 * 80) return;
  int i = idx % 80;
  int row = idx / 80;
  int t = row >> 2, b = row & 3;
  xs[idx] = sig[((size_t)b * 200 + t) * 80 + i];
}

// embeddings -> Xdec (780,128), rows = s*12 + p*4 + b; s==0 uses SOS=1
__global__ void embed_kernel(const int* __restrict__ tg, const float* __restrict__ ec,
                             const float* __restrict__ ev, const float* __restrict__ et,
                             float* __restrict__ X)
{
  int idx = blockIdx.x * 256 + threadIdx.x;
  if (idx >= 780 * 128) return;
  int e = idx & 127;
  int row = idx >> 7;
  int b = row & 3;
  int p = (row >> 2) % 3;
  int s = row / 12;
  int tok = (s == 0) ? 1 : tg[((size_t)b * 64 + (s - 1)) * 3 + p];
  const float* tab = (p == 0) ? ec : (p == 1) ? ev : et;
  X[idx] = tab[(size_t)tok * 128 + e];
}

// logits[b,t,u,v] = encW[t*4+b, v] + decW[u*12+p*4+b, v] + bias[v]
__global__ void joint_kernel(const float* __restrict__ encW, const float* __restrict__ decW,
                             const float* __restrict__ bias, float* __restrict__ out,
                             int Nv, int p, int total)
{
  int idx = blockIdx.x * 256 + threadIdx.x;
  if (idx >= total) return;
  int v = idx % Nv;
  int r = idx / Nv;
  int u = r % 65; r /= 65;
  int t = r % 200;
  int b = r / 200;
  out[idx] = encW[(size_t)((t << 2) + b) * Nv + v]
           + decW[(size_t)(u * 12 + (p << 2) + b) * Nv + v]
           + bias[v];
}

extern "C" void kernel_launch(void* const* d_in, const int* in_sizes, int n_in,
                              void* d_out, int out_size, void* d_ws, size_t ws_size,
                              hipStream_t stream)
{
  (void)in_sizes; (void)n_in; (void)out_size; (void)ws_size;
  const float* signal  = (const float*)d_in[0];
  const int*   targets = (const int*)  d_in[1];
  const float* enc_Wi0 = (const float*)d_in[2];
  const float* enc_Wh0 = (const float*)d_in[3];
  const float* enc_b0  = (const float*)d_in[4];
  const float* enc_Wi  = (const float*)d_in[5];
  const float* enc_Wh  = (const float*)d_in[6];
  const float* enc_b   = (const float*)d_in[7];
  const float* emb_c   = (const float*)d_in[8];
  const float* emb_v   = (const float*)d_in[9];
  const float* emb_t   = (const float*)d_in[10];
  const float* dec_Wi0 = (const float*)d_in[11];
  const float* dec_Wh0 = (const float*)d_in[12];
  const float* dec_b0  = (const float*)d_in[13];
  const float* dec_Wi1 = (const float*)d_in[14];
  const float* dec_Wh1 = (const float*)d_in[15];
  const float* dec_b1  = (const float*)d_in[16];
  const float* Wc = (const float*)d_in[17];
  const float* bc = (const float*)d_in[18];
  const float* Wv = (const float*)d_in[19];
  const float* bv = (const float*)d_in[20];
  const float* Wt = (const float*)d_in[21];
  const float* bt = (const float*)d_in[22];
  float* out = (float*)d_out;

  // workspace layout (floats); total ~24.5 MB
  float* ws   = (float*)d_ws;
  float* G    = ws;              // 2*800*2048 = 3,276,800 (reused for dec G0: 2*784*2048)
  float* bufA = ws + 3276800;    // 800*1024
  float* bufB = ws + 4096000;    // 800*1024
  float* misc = ws + 4915200;    // xs0 (64,000) then Xdec (100,352)
  float* y1   = ws + 5015552;    // 784*1024
  float* encW = ws + 5818368;    // 800*(32+160+8)
  float* decW = ws + 5978368;    // 784*(32+160+8)
  float* xs0  = misc;
  float* Xdec = misc;
  float* encWc = encW, *encWv = encW + 25600,  *encWt = encW + 153600;
  float* decWc = decW, *decWv = decW + 25088,  *decWt = decW + 150528;

  // ---- encoder ----
  transpose_signal_kernel<<<250, 256, 0, stream>>>(signal, xs0);
  for (int d = 0; d < 2; ++d)
    gemm_wmma_kernel<<<800, 256, 0, stream>>>(xs0, enc_Wi0 + (size_t)d * 2048 * 80,
        enc_b0 + d * 2048, G + (size_t)d * 800 * 2048, 800, 2048, 80, 2048);
  enc_rec_kernel<<<2, 1024, 0, stream>>>(G, enc_Wh0, bufA, 200);

  float* cur = bufA; float* nxt = bufB;
  for (int l = 1; l < 5; ++l) {
    for (int d = 0; d < 2; ++d)
      gemm_wmma_kernel<<<800, 256, 0, stream>>>(cur,
          enc_Wi + (size_t)((l - 1) * 2 + d) * 2048 * 1024,
          enc_b  + (size_t)((l - 1) * 2 + d) * 2048,
          G + (size_t)d * 800 * 2048, 800, 2048, 1024, 2048);
    enc_rec_kernel<<<2, 1024, 0, stream>>>(G, enc_Wh + (size_t)(l - 1) * 2 * 2048 * 512,
                                           nxt, 200);
    float* tmp = cur; cur = nxt; nxt = tmp;
  }
  const float* out_enc = cur;    // = bufA after 4 swaps

  // ---- decoder ----
  embed_kernel<<<390, 256, 0, stream>>>(targets, emb_c, emb_v, emb_t, Xdec);
  for (int d = 0; d < 2; ++d)
    gemm_wmma_kernel<<<784, 256, 0, stream>>>(Xdec, dec_Wi0 + (size_t)d * 2048 * 128,
        dec_b0 + d * 2048, G + (size_t)d * 784 * 2048, 780, 2048, 128, 2048);
  (void)hipFuncSetAttribute(reinterpret_cast<const void*>(dec_rec_kernel),
                            hipFuncAttributeMaxDynamicSharedMemorySize, 147456);
  dec_rec_kernel<<<1, 1024, 147456, stream>>>(G, dec_Wh0, dec_Wi1, dec_Wh1, dec_b1, y1);

  // ---- factorized joint heads: enc@W.T and dec@W.T separately ----
  gemm_wmma_kernel<<<13, 256, 0, stream>>>(out_enc, Wc, nullptr, encWc, 800, 32, 1024, 32);
  gemm_wmma_kernel<<<63, 256, 0, stream>>>(out_enc, Wv, nullptr, encWv, 800, 160, 1024, 160);
  gemm_wmma_kernel<<< 7, 256, 0, stream>>>(out_enc, Wt, nullptr, encWt, 800, 8, 1024, 8);
  gemm_wmma_kernel<<<13, 256, 0, stream>>>(y1, Wc, nullptr, decWc, 780, 32, 1024, 32);
  gemm_wmma_kernel<<<62, 256, 0, stream>>>(y1, Wv, nullptr, decWv, 780, 160, 1024, 160);
  gemm_wmma_kernel<<< 7, 256, 0, stream>>>(y1, Wt, nullptr, decWt, 780, 8, 1024, 8);

  // ---- broadcast add (HBM streaming, ~42 MB @ 23.3 TB/s) ----
  joint_kernel<<< 6500, 256, 0, stream>>>(encWc, decWc, bc, out,           32, 0, 1664000);
  joint_kernel<<<32500, 256, 0, stream>>>(encWv, decWv, bv, out + 1664000, 160, 1, 8320000);
  joint_kernel<<< 1625, 256, 0, stream>>>(encWt, decWt, bt, out + 9984000,  8, 2, 416000);
}